// DetectionLoss_42958262894822
// MI455X (gfx1250) — compile-verified
//
#include <hip/hip_runtime.h>
#include <math.h>

#define BB 8
#define SS 4
#define NN 8192
#define MM 50
#define MPAD 64
#define KTOP 10
#define COST_THRESH2 2500.0f   // 50^2, compare in squared-distance domain

typedef float v2f __attribute__((ext_vector_type(2)));
typedef float v8f __attribute__((ext_vector_type(8)));

#define ACC_CENTER 0
#define ACC_S1 1
#define ACC_S2 2
#define ACC_NPOS 3
#define ACC_UNM 4

__device__ __forceinline__ float softplusf(float x) {
    return fmaxf(x, 0.f) + log1pf(expf(-fabsf(x)));
}
__device__ __forceinline__ float smooth_l1(float x) {
    float a = fabsf(x);
    return (a < 1.f) ? (0.5f * a * a) : (a - 0.5f);
}

__global__ void init_kernel(int* assign, float* accum, int* flags) {
    int i = blockIdx.x * blockDim.x + threadIdx.x;
    if (i < BB * SS * NN) assign[i] = -1;
    if (i < 8)  accum[i] = 0.f;
    if (i < 32) flags[i] = 0;
}

// One block per (b,s,gt-tile-of-16). 8 waves; each wave scans N/8 preds for
// the 16 GT columns of its tile. Squared distances come straight out of
// V_WMMA_F32_16X16X4_F32 via  A=[x,y,z,|p|^2], B=[-2gx,-2gy,-2gz,1], C=0,
// then + |g|^2 (lane-constant). Each lane keeps a register top-10 (squared)
// for its GT column; 16 per-wave sorted lists merge through LDS.
__global__ void topk_kernel(const float* __restrict__ pred_boxes,
                            const float* __restrict__ gt_boxes,
                            float* __restrict__ tk_dist,
                            int*   __restrict__ tk_idx) {
    const int blk  = blockIdx.x;      // ((b*S+s)*4 + tile)
    const int tile = blk & 3;
    const int bs   = blk >> 2;        // b*S+s
    const int b    = bs / SS;

    const int tid  = threadIdx.x;
    const int wave = tid >> 5;
    const int lane = tid & 31;
    const int col  = lane & 15;
    const int gt   = tile * 16 + col;

    __shared__ float lds_d[16][16][KTOP];
    __shared__ int   lds_i[16][16][KTOP];

    // GT column data (lane-constant across the N loop)
    float gx = 0.f, gy = 0.f, gz = 0.f;
    if (gt < MM) {
        const float* gp = gt_boxes + (size_t)(b * MM + gt) * 7;
        gx = gp[0]; gy = gp[1]; gz = gp[2];
    }
    const float gm2 = gx * gx + gy * gy + gz * gz;

    // B-frag (4x16): lanes 0-15 hold K0,K1 = -2gx,-2gy ; lanes 16-31 K2,K3 = -2gz, 1
    v2f bfrag;
    if (lane < 16) { bfrag.x = -2.f * gx; bfrag.y = -2.f * gy; }
    else           { bfrag.x = -2.f * gz; bfrag.y = 1.f;       }

    float td[KTOP]; int ti[KTOP];
#pragma unroll
    for (int k = 0; k < KTOP; ++k) { td[k] = 3.4e38f; ti[k] = 0; }

    const int   hi    = (lane < 16) ? 0 : 8;
    const int   row16 = lane & 15;
    const float* pbase = pred_boxes + (size_t)bs * NN * 7;

    const int n_start = wave * (NN / 8);
    for (int n0 = n_start; n0 < n_start + NN / 8; n0 += 16) {
        // Every lane loads x,y,z of its row (rows 0-15 duplicated across halves)
        const float* pb = pbase + (size_t)(n0 + row16) * 7;
        float px = pb[0], py = pb[1], pz = pb[2];
        __builtin_prefetch(pb + 16 * 7, 0, 0);

        // A-frag (16x4): lanes 0-15: K0,K1 = x,y ; lanes 16-31: K2,K3 = z,|p|^2
        v2f af;
        if (lane < 16) { af.x = px; af.y = py; }
        else           { af.x = pz; af.y = px * px + py * py + pz * pz; }

        v8f cf = {};   // C = 0
        // D[row][col] = |p|^2 - 2 p.g   (per lane: one gt col, 8 pred rows)
        v8f d2 = __builtin_amdgcn_wmma_f32_16x16x4_f32(
            false, af, false, bfrag, (short)0, cf, false, false);

#pragma unroll
        for (int r = 0; r < 8; ++r) {
            float dist2 = d2[r] + gm2;          // squared distance
            int   idx   = n0 + r + hi;
            if (dist2 < td[KTOP - 1]) {
                td[KTOP - 1] = dist2; ti[KTOP - 1] = idx;
#pragma unroll
                for (int k = KTOP - 1; k > 0; --k) {
                    if (td[k] < td[k - 1]) {
                        float t0 = td[k]; td[k] = td[k - 1]; td[k - 1] = t0;
                        int   t1 = ti[k]; ti[k] = ti[k - 1]; ti[k - 1] = t1;
                    } else break;
                }
            }
        }
    }

    // dump 16 sorted lists per GT column (8 waves x 2 half-lanes)
    const int list = wave * 2 + ((lane < 16) ? 0 : 1);
#pragma unroll
    for (int k = 0; k < KTOP; ++k) { lds_d[col][list][k] = td[k]; lds_i[col][list][k] = ti[k]; }
    __syncthreads();

    if (tid < 16) {
        float md[KTOP]; int mi[KTOP];
#pragma unroll
        for (int k = 0; k < KTOP; ++k) { md[k] = 3.4e38f; mi[k] = 0; }
        for (int L = 0; L < 16; ++L) {
            for (int k = 0; k < KTOP; ++k) {
                float dv = lds_d[tid][L][k];
                if (dv >= md[KTOP - 1]) break;   // lists are sorted ascending
                int iv = lds_i[tid][L][k];
                md[KTOP - 1] = dv; mi[KTOP - 1] = iv;
                for (int q = KTOP - 1; q > 0; --q) {
                    if (md[q] < md[q - 1]) {
                        float t0 = md[q]; md[q] = md[q - 1]; md[q - 1] = t0;
                        int   t1 = mi[q]; mi[q] = mi[q - 1]; mi[q - 1] = t1;
                    } else break;
                }
            }
        }
        const int m = tile * 16 + tid;
        if (m < MM) {
            float* od = tk_dist + ((size_t)bs * MPAD + m) * KTOP;
            int*   oi = tk_idx  + ((size_t)bs * MPAD + m) * KTOP;
            for (int k = 0; k < KTOP; ++k) { od[k] = md[k]; oi[k] = mi[k]; }
        }
    }
}

// One block per (b,s); thread m handles GT m: scatter-max assignment for the
// <thresh picks, unmatched-GT soft term from the 5 nearest (sqrt only here).
__global__ void assign_kernel(const float* __restrict__ tk_dist,
                              const int*   __restrict__ tk_idx,
                              const int*   __restrict__ n_gt,
                              const float* __restrict__ scores,
                              int* assign, float* accum, int* flags) {
    int bs = blockIdx.x;
    int b  = bs / SS;
    int m  = threadIdx.x;
    if (m >= MM) return;
    if (m >= n_gt[b]) return;
    const float* d = tk_dist + ((size_t)bs * MPAD + m) * KTOP;
    const int*   x = tk_idx  + ((size_t)bs * MPAD + m) * KTOP;
    bool matched = false;
    for (int k = 0; k < KTOP; ++k) {
        if (d[k] < COST_THRESH2) {
            matched = true;
            atomicMax(&assign[(size_t)bs * NN + x[k]], m);
        }
    }
    if (!matched) {
        float acc = 0.f;
        for (int k = 0; k < 5; ++k) {
            float dist = sqrtf(fmaxf(d[k], 0.f));
            acc += expf(-0.5f * dist) * (1.f - scores[(size_t)bs * NN + x[k]]);
        }
        atomicAdd(&accum[ACC_UNM], acc * (1.f / 5.f));
        atomicOr(&flags[bs], 1);
    }
}

// Per-prediction smooth-L1 center loss + BCE partial sums, block-reduced.
__global__ void loss_main_kernel(const float* __restrict__ pred_boxes,
                                 const float* __restrict__ scores,
                                 const float* __restrict__ gt_boxes,
                                 const int*   __restrict__ assign,
                                 float* accum) {
    int i = blockIdx.x * blockDim.x + threadIdx.x;   // exactly B*S*N threads
    int b = i / (SS * NN);
    int a = assign[i];
    float x  = scores[i];
    float s1 = 0.f, s2 = 0.f, np = 0.f, cs = 0.f;
    if (a >= 0) {
        np = 1.f;
        s1 = softplusf(-x);
        const float* pp = pred_boxes + (size_t)i * 7;
        const float* gp = gt_boxes + ((size_t)b * MM + a) * 7;
        cs = smooth_l1(pp[0] - gp[0]) + smooth_l1(pp[1] - gp[1]) + smooth_l1(pp[2] - gp[2]);
    } else {
        s2 = softplusf(x);
    }
    __shared__ float red[4][256];
    int t = threadIdx.x;
    red[0][t] = cs; red[1][t] = s1; red[2][t] = s2; red[3][t] = np;
    __syncthreads();
    for (int off = 128; off > 0; off >>= 1) {
        if (t < off) {
            red[0][t] += red[0][t + off];
            red[1][t] += red[1][t + off];
            red[2][t] += red[2][t + off];
            red[3][t] += red[3][t + off];
        }
        __syncthreads();
    }
    if (t == 0) {
        atomicAdd(&accum[ACC_CENTER], red[0][0]);
        atomicAdd(&accum[ACC_S1],     red[1][0]);
        atomicAdd(&accum[ACC_S2],     red[2][0]);
        atomicAdd(&accum[ACC_NPOS],   red[3][0]);
    }
}

__global__ void finalize_kernel(const float* accum, const int* flags,
                                const float* w_center, const float* w_obj,
                                const float* w_unm, float* out) {
    float n_pos = accum[ACC_NPOS];
    float loss_center = accum[ACC_CENTER] / fmaxf(n_pos * 3.f, 1.f);
    const float total = (float)(BB * SS * NN);
    float n_neg = total - n_pos;
    float pw = fminf(10.f, n_neg / fmaxf(n_pos, 1.f));
    float loss_obj = (pw * accum[ACC_S1] + accum[ACC_S2]) / total;
    float ne = 0.f;
    for (int i = 0; i < BB * SS; ++i) ne += (flags[i] ? 1.f : 0.f);
    float loss_unm = accum[ACC_UNM] / fmaxf(ne, 1.f);
    out[0] = loss_center * w_center[0] + loss_obj * w_obj[0] + loss_unm * w_unm[0];
}

extern "C" void kernel_launch(void* const* d_in, const int* in_sizes, int n_in,
                              void* d_out, int out_size, void* d_ws, size_t ws_size,
                              hipStream_t stream) {
    const float* pred_boxes  = (const float*)d_in[0];
    const float* pred_scores = (const float*)d_in[2];
    const float* gt_boxes    = (const float*)d_in[3];
    const int*   n_gt        = (const int*)d_in[5];
    const float* w_center    = (const float*)d_in[6];
    const float* w_obj       = (const float*)d_in[7];
    const float* w_unm       = (const float*)d_in[8];
    float* out = (float*)d_out;

    char* ws = (char*)d_ws;
    float* accum   = (float*)ws;                         // 8 floats
    int*   flags   = (int*)(ws + 64);                    // 32 ints
    float* tk_dist = (float*)(ws + 256);                 // B*S*MPAD*KTOP floats = 80 KB
    int*   tk_idx  = (int*)(ws + 256 + BB * SS * MPAD * KTOP * 4);
    int*   assign  = (int*)(ws + 256 + 2 * BB * SS * MPAD * KTOP * 4);  // B*S*N ints = 1 MB

    init_kernel<<<(BB * SS * NN + 255) / 256, 256, 0, stream>>>(assign, accum, flags);
    topk_kernel<<<BB * SS * 4, 256, 0, stream>>>(pred_boxes, gt_boxes, tk_dist, tk_idx);
    assign_kernel<<<BB * SS, 64, 0, stream>>>(tk_dist, tk_idx, n_gt, pred_scores,
                                              assign, accum, flags);
    loss_main_kernel<<<(BB * SS * NN) / 256, 256, 0, stream>>>(pred_boxes, pred_scores,
                                                               gt_boxes, assign, accum);
    finalize_kernel<<<1, 1, 0, stream>>>(accum, flags, w_center, w_obj, w_unm, out);
}